// VariationalGCNEncoder_54924041781476
// MI455X (gfx1250) — compile-verified
//
#include <hip/hip_runtime.h>

typedef __attribute__((ext_vector_type(2))) float v2f;
typedef __attribute__((ext_vector_type(8))) float v8f;

// ---------------------------------------------------------------------------
// Degree / normalization
// ---------------------------------------------------------------------------
__global__ void deg_init_k(float* __restrict__ deg, int n) {
  int i = blockIdx.x * blockDim.x + threadIdx.x;
  if (i < n) deg[i] = 1.0f;  // self-loop contributes 1
}

__global__ void deg_count_k(const int* __restrict__ dst, float* __restrict__ deg, int e) {
  int i = blockIdx.x * blockDim.x + threadIdx.x;
  if (i < e) atomicAdd(&deg[dst[i]], 1.0f);
}

__global__ void dinv_k(const float* __restrict__ deg, float* __restrict__ dinv, int n) {
  int i = blockIdx.x * blockDim.x + threadIdx.x;
  if (i < n) dinv[i] = rsqrtf(deg[i]);  // deg >= 1 always (self loops)
}

// ---------------------------------------------------------------------------
// Wave-level f32 WMMA GEMM: Z[M x NC] = X[M x K] @ W[K x NC], row-major.
// Compile-time K/NC -> all strides are immediate offsets; A is indexed off a
// single base (keeps address-space inference -> global_load_b64, and the
// per-tile t*16*K / k terms fold into the 24-bit instruction offset).
// W is staged in LDS once per block (padded stride kills bank conflicts).
// Each wave owns MT consecutive 16x16 output tiles: one B fragment feeds MT
// independent V_WMMA_F32_16X16X4_F32 chains (hides XDL latency, ISA 7.12.1).
//   A 16x4 layout (ISA 7.12.2): lanes 0-15 hold M=lane, K={k,k+1};
//                               lanes 16-31 hold M=lane-16, K={k+2,k+3}.
//   B 4x16 (row-striped):       lanes 0-15 col=lane, K rows {k,k+1};
//                               lanes 16-31 col=lane-16, K rows {k+2,k+3}.
//   C/D 16x16: VGPR j -> row j (lanes 0-15) / row j+8 (lanes 16-31).
// Requires: gridDim.x * MT * 16 == M, blockDim.x == 32 * (NC/16).
// ---------------------------------------------------------------------------
template <int K, int NC, int MT>
__global__ __launch_bounds__(128) void gemm_wmma_f32_k(
    const float* __restrict__ X, const float* __restrict__ W,
    float* __restrict__ Z) {
  constexpr int LSTRIDE = NC + 16;  // pad: lanes 0-15 and 16-31 hit disjoint banks
  __shared__ float wlds[K * LSTRIDE];

  // Cooperatively stage W (K x NC) into LDS, 16B-wide.
  {
    const float4* wsrc = (const float4*)W;
    for (int idx = threadIdx.x; idx < K * NC / 4; idx += blockDim.x) {
      int row = idx / (NC / 4);
      int c4  = idx % (NC / 4);
      float4 v = wsrc[idx];
      float* p = &wlds[row * LSTRIDE + c4 * 4];
      p[0] = v.x; p[1] = v.y; p[2] = v.z; p[3] = v.w;
    }
  }
  __syncthreads();

  const int lane = threadIdx.x & 31;
  const int wave = threadIdx.x >> 5;

  const int m0 = blockIdx.x * (16 * MT);
  const int n0 = wave * 16;  // blockDim 128 = 4 waves cover NC = 64

  const int hi   = lane >> 4;
  const int l16  = lane & 15;
  const int koff = hi * 2;   // K sub-offset within the 4-wide slab
  const int bcol = n0 + l16;

  // Single A base; per-tile/per-step displacement is a compile-time immediate.
  const float* abase = X + (size_t)(m0 + l16) * K + koff;
  const float* bp = &wlds[koff * LSTRIDE + bcol];

  v8f acc[MT] = {};

  for (int k = 0; k < K; k += 4) {
    v2f b;
    b.x = bp[0];          // W[k+koff  ][bcol]
    b.y = bp[LSTRIDE];    // W[k+koff+1][bcol]
    bp += 4 * LSTRIDE;
#pragma unroll
    for (int t = 0; t < MT; ++t) {
      if (t == 0) __builtin_prefetch(abase + k + 16, 0, 1);  // next line of A
      v2f a = *(const v2f*)(abase + t * (16 * K) + k);
      acc[t] = __builtin_amdgcn_wmma_f32_16x16x4_f32(
          /*neg_a=*/false, a, /*neg_b=*/false, b,
          /*c_mod=*/(short)0, acc[t], /*reuse_a=*/false, /*reuse_b=*/false);
    }
  }

  // Epilogue: one base address, all 8*MT stores use immediate offsets.
  float* zbase = Z + (size_t)(m0 + hi * 8) * NC + bcol;
#pragma unroll
  for (int t = 0; t < MT; ++t) {
#pragma unroll
    for (int j = 0; j < 8; ++j) {
      zbase[(t * 16 + j) * NC] = acc[t][j];
    }
  }
}

// ---------------------------------------------------------------------------
// Aggregation helpers (C = 64 channels fixed)
// ---------------------------------------------------------------------------
// acc[i][c] = dinv[i]^2 * Z[i][c]   (self-loop term initializes accumulator)
__global__ void self_init_k(const float* __restrict__ Z,
                            const float* __restrict__ dinv,
                            float* __restrict__ acc, int total) {
  int idx = blockIdx.x * blockDim.x + threadIdx.x;
  if (idx < total) {
    int i = idx >> 6;
    float d = dinv[i];
    acc[idx] = d * d * Z[idx];
  }
}

// acc[dst][c] += dinv[src]*dinv[dst] * Z[src][c]  over all edges
__global__ void scatter_k(const float* __restrict__ Z,
                          const float* __restrict__ dinv,
                          const int* __restrict__ src,
                          const int* __restrict__ dst,
                          float* __restrict__ acc, int total /* E*64 */) {
  int idx = blockIdx.x * blockDim.x + threadIdx.x;
  if (idx < total) {
    int e = idx >> 6;
    int c = idx & 63;
    int s = src[e];
    int d = dst[e];
    float w = dinv[s] * dinv[d];
    atomicAdd(&acc[(size_t)d * 64 + c], w * Z[(size_t)s * 64 + c]);
  }
}

// h[i][c] = relu(acc[i][c] + b[c])
__global__ void relu_bias_k(const float* __restrict__ acc,
                            const float* __restrict__ b,
                            float* __restrict__ h, int total) {
  int idx = blockIdx.x * blockDim.x + threadIdx.x;
  if (idx < total) {
    int c = idx & 63;
    h[idx] = fmaxf(acc[idx] + b[c], 0.0f);
  }
}

// Wcat[k][j] = j<32 ? W_mu[k][j] : W_ls[k][j-32]   (64 x 64)
__global__ void build_wcat_k(const float* __restrict__ Wmu,
                             const float* __restrict__ Wls,
                             float* __restrict__ Wcat) {
  int idx = blockIdx.x * blockDim.x + threadIdx.x;
  if (idx < 64 * 64) {
    int k = idx >> 6;
    int j = idx & 63;
    Wcat[idx] = (j < 32) ? Wmu[k * 32 + j] : Wls[k * 32 + (j - 32)];
  }
}

// Layer-2 self term + bias, written straight into d_out (mu || logstd layout).
__global__ void self_init_out_k(const float* __restrict__ Z2,
                                const float* __restrict__ dinv,
                                const float* __restrict__ bmu,
                                const float* __restrict__ bls,
                                float* __restrict__ out, int N) {
  int idx = blockIdx.x * blockDim.x + threadIdx.x;
  int total = N * 64;
  if (idx < total) {
    int i = idx >> 6;
    int c = idx & 63;
    float d = dinv[i];
    float v = d * d * Z2[idx];
    if (c < 32) {
      out[(size_t)i * 32 + c] = v + bmu[c];
    } else {
      out[(size_t)N * 32 + (size_t)i * 32 + (c - 32)] = v + bls[c - 32];
    }
  }
}

// Layer-2 edge scatter directly into d_out.
__global__ void scatter_out_k(const float* __restrict__ Z2,
                              const float* __restrict__ dinv,
                              const int* __restrict__ src,
                              const int* __restrict__ dst,
                              float* __restrict__ out, int N, int total /* E*64 */) {
  int idx = blockIdx.x * blockDim.x + threadIdx.x;
  if (idx < total) {
    int e = idx >> 6;
    int c = idx & 63;
    int s = src[e];
    int d = dst[e];
    float w = dinv[s] * dinv[d];
    float v = w * Z2[(size_t)s * 64 + c];
    float* p = (c < 32) ? (out + (size_t)d * 32 + c)
                        : (out + (size_t)N * 32 + (size_t)d * 32 + (c - 32));
    atomicAdd(p, v);
  }
}

// ---------------------------------------------------------------------------
// Launch
// ---------------------------------------------------------------------------
extern "C" void kernel_launch(void* const* d_in, const int* in_sizes, int n_in,
                              void* d_out, int out_size, void* d_ws, size_t ws_size,
                              hipStream_t stream) {
  (void)in_sizes; (void)n_in; (void)out_size; (void)ws_size;

  constexpr int N  = 50000;
  constexpr int E  = 800000;
  constexpr int IN = 256;
  constexpr int HC = 64;   // hidden channels == scatter channel count
  constexpr int MT = 5;    // M-tiles per wave; 3125 tiles = 625 blocks * 5

  const float* x    = (const float*)d_in[0];   // [N, 256]
  const float* W_in = (const float*)d_in[1];   // [256, 64]
  const float* b_in = (const float*)d_in[2];   // [64]
  const float* W_mu = (const float*)d_in[3];   // [64, 32]
  const float* b_mu = (const float*)d_in[4];   // [32]
  const float* W_ls = (const float*)d_in[5];   // [64, 32]
  const float* b_ls = (const float*)d_in[6];   // [32]
  const int*   eidx = (const int*)d_in[7];     // [2, E]: row 0 = src, row 1 = dst
  const int* src = eidx;
  const int* dst = eidx + E;

  float* out = (float*)d_out;                  // mu [N,32] || logstd [N,32]

  // Workspace layout (floats): deg | dinv | bufA | bufB | Wcat  (~26 MB)
  float* ws   = (float*)d_ws;
  float* deg  = ws;
  float* dinv = deg  + N;
  float* bufA = dinv + N;                      // Z1, later h
  float* bufB = bufA + (size_t)N * HC;         // acc1, later Z2
  float* wcat = bufB + (size_t)N * HC;         // 64x64 fused [W_mu | W_ls]

  const int TB = 256;
  const int NC = N * HC;                       // 3.2M
  const int EC = E * HC;                       // 51.2M

  // 1) degrees + normalization
  deg_init_k<<<(N + TB - 1) / TB, TB, 0, stream>>>(deg, N);
  deg_count_k<<<(E + TB - 1) / TB, TB, 0, stream>>>(dst, deg, E);
  dinv_k<<<(N + TB - 1) / TB, TB, 0, stream>>>(deg, dinv, N);

  // 2) Z1 = X @ W_in  (WMMA f32, LDS-staged B, 5 M-tiles/wave)
  gemm_wmma_f32_k<IN, HC, MT>
      <<<dim3(N / (16 * MT)), dim3(128), 0, stream>>>(x, W_in, bufA);

  // 3) Aggregate layer 1: acc = dinv^2*Z1 ; acc[dst] += norm*Z1[src] ; h = relu(acc+b)
  self_init_k<<<(NC + TB - 1) / TB, TB, 0, stream>>>(bufA, dinv, bufB, NC);
  scatter_k<<<(EC + TB - 1) / TB, TB, 0, stream>>>(bufA, dinv, src, dst, bufB, EC);
  relu_bias_k<<<(NC + TB - 1) / TB, TB, 0, stream>>>(bufB, b_in, bufA, NC);  // h -> bufA

  // 4) Z2 = h @ [W_mu | W_ls]  (single fused 64x64 GEMM)
  build_wcat_k<<<(64 * 64 + TB - 1) / TB, TB, 0, stream>>>(W_mu, W_ls, wcat);
  gemm_wmma_f32_k<HC, HC, MT>
      <<<dim3(N / (16 * MT)), dim3(128), 0, stream>>>(bufA, wcat, bufB);  // Z2 -> bufB

  // 5) Aggregate layer 2 directly into d_out (mu || logstd)
  self_init_out_k<<<(NC + TB - 1) / TB, TB, 0, stream>>>(bufB, dinv, b_mu, b_ls, out, N);
  scatter_out_k<<<(EC + TB - 1) / TB, TB, 0, stream>>>(bufB, dinv, src, dst, out, N, EC);
}